// NodeLevelInnerProductDecoder_58299886076485
// MI455X (gfx1250) — compile-verified
//
#include <hip/hip_runtime.h>
#include <hip/hip_bf16.h>

typedef __attribute__((ext_vector_type(2))) float v2f;
typedef __attribute__((ext_vector_type(8))) float v8f;

#define NGRAPH   64
#define DDIM     128
#define MNODES   508
#define NVIEWS   4

// One wave32 per 32x32 macro-tile of one graph's Gram matrix.
// D = A x B + C with V_WMMA_F32_16X16X4_F32, accumulating over K = 128.
// Loads are unconditional (row index clamped into the valid range) and
// zero-padding is applied by scaling the fragment with a 0/1 mask, so the
// inner loop is straight-line: no exec-mask save/restore around loads.
__global__ __launch_bounds__(32)
void NodeLevelInnerProductDecoder_gram_wmma(const float* __restrict__ z,
                                            const long long* __restrict__ counts,
                                            float* __restrict__ out)
{
    const int lane = threadIdx.x;        // 0..31
    const int tn   = blockIdx.x;         // col macro-tile 0..15 (32 cols each)
    const int tm   = blockIdx.y;         // row macro-tile 0..15
    const int b    = blockIdx.z;         // graph 0..63

    // Uniform (scalar) prefix sum over int64 node counts -> row offset of graph b.
    long long off = 0;
    for (int j = 0; j < b; ++j) off += counts[j];
    const int nb = (int)counts[b];       // 256..508
    const float* __restrict__ zb = z + off * DDIM;

    // WMMA f32 16x16x4 fragment layout (wave32):
    //   A (16x4, MxK): lane L holds row M = L&15; VGPR0 = K = 2*(L>>4), VGPR1 = K+1.
    //   B (4x16, KxN) for the Gram is A^T, i.e. same per-lane load with the column index.
    const int rsub = lane & 15;
    const int kgrp = (lane >> 4) << 1;   // 0 or 2

    const int m0 = tm * 32 + rsub;       // A tile rows (two 16-row tiles)
    const int m1 = m0 + 16;
    const int n0 = tn * 32 + rsub;       // B tile cols (two 16-col tiles)
    const int n1 = n0 + 16;

    // Zero-pad masks as 0.0/1.0 scales (rows >= node_count contribute zero).
    const float s0 = (m0 < nb) ? 1.0f : 0.0f;
    const float s1 = (m1 < nb) ? 1.0f : 0.0f;
    const float s2 = (n0 < nb) ? 1.0f : 0.0f;
    const float s3 = (n1 < nb) ? 1.0f : 0.0f;

    // Clamp row indices so every load is in-bounds (nb >= 256 > any clamp target).
    const int last = nb - 1;
    const int m0c = m0 < last ? m0 : last;
    const int m1c = m1 < last ? m1 : last;
    const int n0c = n0 < last ? n0 : last;
    const int n1c = n1 < last ? n1 : last;

    const float* __restrict__ pA0 = zb + (long long)m0c * DDIM + kgrp;
    const float* __restrict__ pA1 = zb + (long long)m1c * DDIM + kgrp;
    const float* __restrict__ pB0 = zb + (long long)n0c * DDIM + kgrp;
    const float* __restrict__ pB1 = zb + (long long)n1c * DDIM + kgrp;

    v8f c00 = {}; v8f c01 = {}; v8f c10 = {}; v8f c11 = {};

#pragma unroll 4
    for (int k = 0; k < DDIM; k += 4) {
        v2f a0 = *(const v2f*)(pA0 + k);   // global_load_b64, unconditional
        v2f a1 = *(const v2f*)(pA1 + k);
        v2f b0 = *(const v2f*)(pB0 + k);
        v2f b1 = *(const v2f*)(pB1 + k);
        a0 *= s0;                          // apply zero-pad mask in VALU
        a1 *= s1;
        b0 *= s2;
        b1 *= s3;

        // 8 args: (neg_a, A, neg_b, B, c_mod, C, reuse_a, reuse_b)
        c00 = __builtin_amdgcn_wmma_f32_16x16x4_f32(false, a0, false, b0, (short)0, c00, false, false);
        c01 = __builtin_amdgcn_wmma_f32_16x16x4_f32(false, a0, false, b1, (short)0, c01, false, false);
        c10 = __builtin_amdgcn_wmma_f32_16x16x4_f32(false, a1, false, b0, (short)0, c10, false, false);
        c11 = __builtin_amdgcn_wmma_f32_16x16x4_f32(false, a1, false, b1, (short)0, c11, false, false);
    }

    // C/D layout: VGPR i -> row (i + 8*(lane>>4)), col (lane & 15) within the 16x16 tile.
    const int rowoff = (lane >> 4) * 8;
    const int coloff = lane & 15;
    const long long graph_base = (long long)b * NVIEWS * MNODES * MNODES;

#pragma unroll
    for (int i = 0; i < 8; ++i) {
        const int rm0 = tm * 32 + rowoff + i;
        const int rm1 = rm0 + 16;
        const int cn0 = tn * 32 + coloff;
        const int cn1 = cn0 + 16;

        const float v00 = c00[i];
        const float v01 = c01[i];
        const float v10 = c10[i];
        const float v11 = c11[i];

        const bool r0ok = rm0 < MNODES;
        const bool r1ok = rm1 < MNODES;
        const bool c0ok = cn0 < MNODES;
        const bool c1ok = cn1 < MNODES;

#pragma unroll
        for (int v = 0; v < NVIEWS; ++v) {
            float* __restrict__ o = out + graph_base + (long long)v * MNODES * MNODES;
            if (r0ok) {
                if (c0ok) o[(long long)rm0 * MNODES + cn0] = v00;
                if (c1ok) o[(long long)rm0 * MNODES + cn1] = v01;
            }
            if (r1ok) {
                if (c0ok) o[(long long)rm1 * MNODES + cn0] = v10;
                if (c1ok) o[(long long)rm1 * MNODES + cn1] = v11;
            }
        }
    }
}

extern "C" void kernel_launch(void* const* d_in, const int* in_sizes, int n_in,
                              void* d_out, int out_size, void* d_ws, size_t ws_size,
                              hipStream_t stream) {
    (void)in_sizes; (void)n_in; (void)out_size; (void)d_ws; (void)ws_size;
    const float*     z      = (const float*)d_in[0];          // [24448, 128] f32
    const long long* counts = (const long long*)d_in[1];      // [64] int64
    float*           out    = (float*)d_out;                  // [64, 4, 508, 508] f32

    dim3 grid(16, 16, NGRAPH);   // (tn, tm, graph)
    dim3 block(32, 1, 1);        // one wave32 per 32x32 output macro-tile
    NodeLevelInnerProductDecoder_gram_wmma<<<grid, block, 0, stream>>>(z, counts, out);
}